// OptionCriticHead_50835232915722
// MI455X (gfx1250) — compile-verified
//
#include <hip/hip_runtime.h>

#define BATCH   4096
#define HDIM    512
#define NOPT    16
#define OUTDIM  128

typedef __attribute__((ext_vector_type(2))) float v2f;
typedef __attribute__((ext_vector_type(8))) float v8f;

#define WMMA_F32(A, B, C) \
    __builtin_amdgcn_wmma_f32_16x16x4_f32(false, (A), false, (B), (short)0, (C), false, false)

// ---------------------------------------------------------------------------
// Kernel 1: bucket batch rows by option index.
// idxlist holds per-option contiguous row lists, each padded to a multiple of
// 16 with -1 so a WMMA row-tile never straddles two options.
// ---------------------------------------------------------------------------
__global__ void oc_bucket_kernel(const int* __restrict__ option,
                                 int* __restrict__ cnt,
                                 int* __restrict__ base,
                                 int* __restrict__ idxlist) {
    __shared__ int scnt[NOPT];
    __shared__ int sbase[NOPT];
    __shared__ int scur[NOPT];
    const int tid = threadIdx.x;

    if (tid < NOPT) { scnt[tid] = 0; scur[tid] = 0; }
    __syncthreads();

    for (int b = tid; b < BATCH; b += blockDim.x)
        atomicAdd(&scnt[option[b]], 1);
    __syncthreads();

    if (tid == 0) {
        int acc = 0;
        for (int o = 0; o < NOPT; ++o) {
            sbase[o] = acc;
            acc += (scnt[o] + 15) & ~15;   // pad each bucket to 16
        }
    }
    __syncthreads();

    if (tid < NOPT) { cnt[tid] = scnt[tid]; base[tid] = sbase[tid]; }

    // Pre-fill every slot (incl. pad) with -1, then scatter real indices.
    for (int i = tid; i < BATCH + NOPT * 16; i += blockDim.x)
        idxlist[i] = -1;
    __syncthreads();

    for (int b = tid; b < BATCH; b += blockDim.x) {
        int o   = option[b];
        int pos = atomicAdd(&scur[o], 1);
        idxlist[sbase[o] + pos] = b;
    }
}

// ---------------------------------------------------------------------------
// Kernel 2: grouped GEMM via V_WMMA_F32_16X16X4_F32.
// grid = (16 options, 8 col-tiles, 2 pair-chunks), block = 128 (4 waves).
// Each wave processes TWO 16-row tiles (M-blocking) sharing each B fragment.
// The k-loop stays rolled (unroll 1) with a TWO-deep software pipeline:
// chunk k+16's loads are issued while chunk k's WMMAs consume registers
// loaded two iterations ago -> long load->use distance, partial-count waits,
// and no t-invariant hoisting / spilling.
//
// B LDS layout: element (k, n) at wlds[(k>>1)*32 + n*2 + (k&1)], so a
// B fragment is one aligned ds_load_b64 at wbase + 16*k,
// wbase = wlds + hi*32 + n*2.
//
// Pad rows (idx < 0) are clamped to row 0 and loaded unconditionally: their
// garbage stays in D rows that are never stored -> branch-free K loop.
// ---------------------------------------------------------------------------
__global__ void oc_gemm_kernel(const float* __restrict__ x,
                               const float* __restrict__ W,
                               const float* __restrict__ bias,
                               const int*   __restrict__ cnt,
                               const int*   __restrict__ base,
                               const int*   __restrict__ idxlist,
                               float* __restrict__ out) {
    const int o       = blockIdx.x;        // option
    const int colbase = blockIdx.y << 4;   // 16-column tile
    const int tid     = threadIdx.x;       // 0..127

    __shared__ float wlds[HDIM * 16];      // 32 KB

    // Stage W[o][:, colbase : colbase+16]  (coalesced 16-lane rows)
    for (int i = tid; i < HDIM * 16; i += 128) {
        int kk = i >> 4;                   // 0..511
        int n  = i & 15;
        float v = W[(size_t)o * (HDIM * OUTDIM) + (size_t)kk * OUTDIM + colbase + n];
        wlds[((kk >> 1) << 5) + (n << 1) + (kk & 1)] = v;
    }
    __syncthreads();

    const int lane = tid & 31;
    const int wave = tid >> 5;             // 0..3
    const int n    = lane & 15;            // N index (and A-row M index)
    const int hi   = lane >> 4;            // lane-half
    const int koff = hi << 1;              // A/B K sub-offset: 0 or 2

    const int   ntiles = (cnt[o] + 15) >> 4;
    const int   npairs = (ntiles + 1) >> 1;
    const int   ibase  = base[o];
    const float bv     = bias[o * OUTDIM + colbase + n];
    const float* wbase = wlds + (hi << 5) + (n << 1);

    for (int p = (blockIdx.z << 2) + wave; p < npairs; p += 8) {
        const int t0 = p << 1;
        const int t1 = t0 + 1;
        // Row indices for A rows M = n (lanes L and L+16 hold the same M).
        const int rowIdx0 = idxlist[ibase + (t0 << 4) + n];
        const int rowIdx1 = (t1 < ntiles) ? idxlist[ibase + (t1 << 4) + n] : -1;
        // Clamp pad rows to row 0: loads stay in-bounds, results unused.
        const float* xr0 = x + (size_t)(rowIdx0 < 0 ? 0 : rowIdx0) * HDIM + koff;
        const float* xr1 = x + (size_t)(rowIdx1 < 0 ? 0 : rowIdx1) * HDIM + koff;

        v8f c0 = {};
        v8f c1 = {};

        // ---- 2-deep software pipeline: prologue loads chunks 0 and 1 ----
        v2f a00_0 = *(const v2f*)(xr0 + 0);
        v2f a10_0 = *(const v2f*)(xr1 + 0);
        v2f b0_0  = *(const v2f*)(wbase + (0 << 4));
        v2f a01_0 = *(const v2f*)(xr0 + 4);
        v2f a11_0 = *(const v2f*)(xr1 + 4);
        v2f b1_0  = *(const v2f*)(wbase + (4 << 4));

        v2f a00_1 = *(const v2f*)(xr0 + 8);
        v2f a10_1 = *(const v2f*)(xr1 + 8);
        v2f b0_1  = *(const v2f*)(wbase + (8 << 4));
        v2f a01_1 = *(const v2f*)(xr0 + 12);
        v2f a11_1 = *(const v2f*)(xr1 + 12);
        v2f b1_1  = *(const v2f*)(wbase + (12 << 4));

#pragma unroll 1
        for (int k = 0; k < HDIM - 16; k += 8) {
            // Issue chunk k+16's loads; consume chunk k (loaded 2 iters ago).
            v2f na00 = *(const v2f*)(xr0 + k + 16);
            v2f na10 = *(const v2f*)(xr1 + k + 16);
            v2f nb0  = *(const v2f*)(wbase + ((k + 16) << 4));
            v2f na01 = *(const v2f*)(xr0 + k + 20);
            v2f na11 = *(const v2f*)(xr1 + k + 20);
            v2f nb1  = *(const v2f*)(wbase + ((k + 20) << 4));

            c0 = WMMA_F32(a00_0, b0_0, c0);
            c1 = WMMA_F32(a10_0, b0_0, c1);
            c0 = WMMA_F32(a01_0, b1_0, c0);
            c1 = WMMA_F32(a11_0, b1_0, c1);

            // Rotate pipeline registers.
            a00_0 = a00_1; a10_0 = a10_1; b0_0 = b0_1;
            a01_0 = a01_1; a11_0 = a11_1; b1_0 = b1_1;
            a00_1 = na00;  a10_1 = na10;  b0_1 = nb0;
            a01_1 = na01;  a11_1 = na11;  b1_1 = nb1;
        }
        // ---- epilogue: last two chunks ----------------------------------
        c0 = WMMA_F32(a00_0, b0_0, c0);
        c1 = WMMA_F32(a10_0, b0_0, c1);
        c0 = WMMA_F32(a01_0, b1_0, c0);
        c1 = WMMA_F32(a11_0, b1_0, c1);

        c0 = WMMA_F32(a00_1, b0_1, c0);
        c1 = WMMA_F32(a10_1, b0_1, c1);
        c0 = WMMA_F32(a01_1, b1_1, c0);
        c1 = WMMA_F32(a11_1, b1_1, c1);

        // D layout: VGPR r -> (M = r + 8*hi, N = n). Scatter real rows back.
#pragma unroll
        for (int r = 0; r < 8; ++r) {
            int m     = r + (hi << 3);
            int ridx0 = __shfl(rowIdx0, m, 32);  // lane m holds row M=m index
            if (ridx0 >= 0)
                out[(size_t)ridx0 * OUTDIM + colbase + n] = c0[r] + bv;
        }
#pragma unroll
        for (int r = 0; r < 8; ++r) {
            int m     = r + (hi << 3);
            int ridx1 = __shfl(rowIdx1, m, 32);
            if (ridx1 >= 0)
                out[(size_t)ridx1 * OUTDIM + colbase + n] = c1[r] + bv;
        }
    }
}

// ---------------------------------------------------------------------------
// Kernel 3: in-place softmax over OUTDIM=128 per row. One wave32 per row,
// 4 values per lane, shfl_xor tree reductions.
// ---------------------------------------------------------------------------
__global__ void oc_softmax_kernel(float* __restrict__ out) {
    const int lane = threadIdx.x & 31;
    const int wave = threadIdx.x >> 5;
    const int row  = (blockIdx.x << 3) + wave;   // 8 waves / block

    float* p = out + (size_t)row * OUTDIM;
    float4 v = *(const float4*)(p + (lane << 2));

    float m = fmaxf(fmaxf(v.x, v.y), fmaxf(v.z, v.w));
#pragma unroll
    for (int off = 16; off > 0; off >>= 1)
        m = fmaxf(m, __shfl_xor(m, off, 32));

    v.x = expf(v.x - m);
    v.y = expf(v.y - m);
    v.z = expf(v.z - m);
    v.w = expf(v.w - m);

    float s = v.x + v.y + v.z + v.w;
#pragma unroll
    for (int off = 16; off > 0; off >>= 1)
        s += __shfl_xor(s, off, 32);

    const float inv = 1.0f / s;
    v.x *= inv; v.y *= inv; v.z *= inv; v.w *= inv;
    *(float4*)(p + (lane << 2)) = v;
}

// ---------------------------------------------------------------------------
extern "C" void kernel_launch(void* const* d_in, const int* in_sizes, int n_in,
                              void* d_out, int out_size, void* d_ws, size_t ws_size,
                              hipStream_t stream) {
    const float* x      = (const float*)d_in[0];
    const int*   option = (const int*)  d_in[1];
    const float* W      = (const float*)d_in[2];
    const float* b      = (const float*)d_in[3];
    float*       out    = (float*)d_out;

    int* wsi     = (int*)d_ws;
    int* cnt     = wsi;            // 16
    int* base    = wsi + 16;       // 16
    int* idxlist = wsi + 32;       // 4096 + 16*16 pad slots

    oc_bucket_kernel<<<1, 1024, 0, stream>>>(option, cnt, base, idxlist);
    oc_gemm_kernel<<<dim3(NOPT, OUTDIM / 16, 2), 128, 0, stream>>>(
        x, W, b, cnt, base, idxlist, out);
    oc_softmax_kernel<<<BATCH / 8, 256, 0, stream>>>(out);
}